// GNNProcessor_58007828300459
// MI455X (gfx1250) — compile-verified
//
#include <hip/hip_runtime.h>

namespace {

constexpr int  kBGraphs  = 32;
constexpr int  kMaxNodes = 2048;
constexpr int  kN        = kBGraphs * kMaxNodes;   // 65536 nodes
constexpr long long kE   = 1LL << 20;              // 1048576 edges
constexpr int  kHid      = 64;
constexpr int  kOut      = 32;

typedef __attribute__((ext_vector_type(2))) float v2f;
typedef __attribute__((ext_vector_type(8))) float v8f;

// Guaranteed native f32 atomic add (non-returning -> STOREcnt, no VGPR dest).
__device__ __forceinline__ void atomAddF32(float* p, float v) {
  asm volatile("global_atomic_add_f32 %0, %1, off"
               :
               : "v"(p), "v"(v)
               : "memory");
}

// ---- degree / normalization -------------------------------------------------
__global__ void k_deg_init(float* __restrict__ deg) {
  int i = blockIdx.x * blockDim.x + threadIdx.x;
  deg[i] = 1.0f;  // self-loop weight
}

__global__ void k_deg_edges(const long long* __restrict__ ei,
                            const float* __restrict__ ew,
                            float* __restrict__ deg) {
  long long e = (long long)blockIdx.x * blockDim.x + threadIdx.x;
  long long col = ei[kE + e];
  atomAddF32(&deg[col], ew[e]);
}

__global__ void k_dinv(float* __restrict__ deg) {
  int i = blockIdx.x * blockDim.x + threadIdx.x;
  float d = deg[i];
  deg[i] = (d > 0.0f) ? rsqrtf(d) : 0.0f;
}

// ---- layer 1: h1 = x @ W1 ; agg1 = h1*dinv^2 + b1 ---------------------------
__global__ void k_layer1(const float* __restrict__ coords,
                         const float* __restrict__ W1,
                         const float* __restrict__ b1,
                         const float* __restrict__ dinv,
                         float* __restrict__ h1,
                         float* __restrict__ agg1) {
  int t = blockIdx.x * blockDim.x + threadIdx.x;   // N * 64 threads
  int n = t >> 6;
  int j = t & 63;
  float c0 = coords[2 * n + 0];
  float c1 = coords[2 * n + 1];
  float h  = fmaf(c0, W1[j], c1 * W1[kHid + j]);
  h1[t] = h;
  float di = dinv[n];
  agg1[t] = fmaf(h, di * di, b1[j]);
}

// ---- edge scatter: agg[col] += h[row] * dinv[row]*ew*dinv[col] --------------
template <int F>
__global__ void k_scatter(const long long* __restrict__ ei,
                          const float* __restrict__ ew,
                          const float* __restrict__ dinv,
                          const float* __restrict__ hsrc,
                          float* __restrict__ agg) {
  constexpr int L = F / 4;  // lanes per edge, float4 each
  long long t = (long long)blockIdx.x * blockDim.x + threadIdx.x;
  long long e = t / L;
  int       q = (int)(t % L);
  long long row = ei[e];
  long long col = ei[kE + e];
  float norm = dinv[row] * ew[e] * dinv[col];
  const float4 hv = *reinterpret_cast<const float4*>(hsrc + row * F + (q << 2));
  float* dst = agg + col * F + (q << 2);
  atomAddF32(dst + 0, hv.x * norm);
  atomAddF32(dst + 1, hv.y * norm);
  atomAddF32(dst + 2, hv.z * norm);
  atomAddF32(dst + 3, hv.w * norm);
}

// ---- layer 2 GEMM via WMMA f32 16x16x4: h2 = relu(agg1) @ W2 ----------------
// Also initializes outb = h2*dinv^2 + b2 (self-loop + bias for layer-2 agg).
__global__ void k_gemm2_wmma(const float* __restrict__ agg1,
                             const float* __restrict__ W2,
                             const float* __restrict__ b2,
                             const float* __restrict__ dinv,
                             float* __restrict__ h2,
                             float* __restrict__ outb) {
  int wave = (blockIdx.x * blockDim.x + threadIdx.x) >> 5;  // M-tile index
  int lane = threadIdx.x & 31;
  int m  = lane & 15;   // row (A) / col (B,C) within tile
  int hh = lane >> 4;   // K-half selector
  long long row0 = (long long)wave * 16;

  const float* Arow = agg1 + (row0 + m) * kHid + 2 * hh;

  v8f c0 = {};  // output cols 0..15
  v8f c1 = {};  // output cols 16..31
#pragma unroll
  for (int kk = 0; kk < 16; ++kk) {
    int k0 = kk * 4;
    // A lane(hh,m): K = k0+2hh, k0+2hh+1  (contiguous float2; relu fused)
    v2f a;
    a.x = fmaxf(Arow[k0 + 0], 0.0f);
    a.y = fmaxf(Arow[k0 + 1], 0.0f);
    int kr = k0 + 2 * hh;
    v2f b0, b1v;
    b0.x  = W2[(kr + 0) * kOut + m];
    b0.y  = W2[(kr + 1) * kOut + m];
    b1v.x = W2[(kr + 0) * kOut + 16 + m];
    b1v.y = W2[(kr + 1) * kOut + 16 + m];
    c0 = __builtin_amdgcn_wmma_f32_16x16x4_f32(false, a, false, b0,
                                               (short)0, c0, false, false);
    c1 = __builtin_amdgcn_wmma_f32_16x16x4_f32(false, a, false, b1v,
                                               (short)0, c1, false, false);
  }

#pragma unroll
  for (int v = 0; v < 8; ++v) {
    long long gn = row0 + v + 8 * hh;   // C/D: lane half selects M=v or v+8
    float di = dinv[gn];
    float s  = di * di;
    float x0 = c0[v];
    float x1 = c1[v];
    h2[gn * kOut + m] = x0;
    h2[gn * kOut + 16 + m] = x1;
    outb[gn * kOut + m]      = fmaf(x0, s, b2[m]);
    outb[gn * kOut + 16 + m] = fmaf(x1, s, b2[16 + m]);
  }
}

// ---- final: relu + mask -----------------------------------------------------
__global__ void k_final(const unsigned char* __restrict__ mask,
                        float* __restrict__ out) {
  int t = blockIdx.x * blockDim.x + threadIdx.x;  // N*32 threads
  int n = t >> 5;
  float v = fmaxf(out[t], 0.0f);
  out[t] = mask[n] ? v : 0.0f;
}

}  // namespace

extern "C" void kernel_launch(void* const* d_in, const int* in_sizes, int n_in,
                              void* d_out, int out_size, void* d_ws, size_t ws_size,
                              hipStream_t stream) {
  const float*         coords = (const float*)d_in[0];
  const long long*     ei     = (const long long*)d_in[1];  // [2, E] int64
  const float*         ew     = (const float*)d_in[2];
  const unsigned char* mask   = (const unsigned char*)d_in[3];
  const float*         W1     = (const float*)d_in[4];
  const float*         b1     = (const float*)d_in[5];
  const float*         W2     = (const float*)d_in[6];
  const float*         b2     = (const float*)d_in[7];
  float*               outb   = (float*)d_out;

  // workspace carve-up: dinv | h1 (reused as h2) | agg1
  float* dinv = (float*)d_ws;
  float* h1   = dinv + kN;
  float* agg1 = h1 + (size_t)kN * kHid;
  float* h2   = h1;  // h1 dead once gemm2 starts

  k_deg_init <<<kN / 256, 256, 0, stream>>>(dinv);
  k_deg_edges<<<(unsigned)(kE / 256), 256, 0, stream>>>(ei, ew, dinv);
  k_dinv     <<<kN / 256, 256, 0, stream>>>(dinv);

  k_layer1   <<<(kN * kHid) / 256, 256, 0, stream>>>(coords, W1, b1, dinv, h1, agg1);
  k_scatter<64><<<(unsigned)((kE * 16) / 256), 256, 0, stream>>>(ei, ew, dinv, h1, agg1);

  k_gemm2_wmma<<<(kN / 16) / 8, 256, 0, stream>>>(agg1, W2, b2, dinv, h2, outb);
  k_scatter<32><<<(unsigned)((kE * 8) / 256), 256, 0, stream>>>(ei, ew, dinv, h2, outb);

  k_final    <<<(kN * kOut) / 256, 256, 0, stream>>>(mask, outb);
}